// INT8BertEmbeddings_76596446757279
// MI455X (gfx1250) — compile-verified
//
#include <hip/hip_runtime.h>

// INT8 BERT embeddings + approx-LayerNorm, MI455X (gfx1250).
// Memory-bound fused gather/dequant/LN: one 192-thread block (6 wave32s) per
// (b,s) row of H=768. All table traffic is 128-bit coalesced; LayerNorm params
// staged via gfx1250 async global->LDS (ASYNCcnt); output streamed with NT
// b128 stores so the 100 MB write stream doesn't evict the L2-resident vocab
// table. Grid is (S, B) so the position id is blockIdx.x — no int division.
//
// Roofline: ~195 MB compulsory HBM traffic (100.7 MB output NT-stream +
// ~93.8 MB word_table first touch; pos/type/ln tables are L2-resident) at
// 23.3 TB/s => ~8.4 us floor. ~0.3 GFLOP total => purely bandwidth-bound;
// WMMA has no role (no GEMM) — the gfx1250 leverage is the memory path.

typedef float f4  __attribute__((ext_vector_type(4)));
typedef int   i4  __attribute__((ext_vector_type(4)));
typedef int   v4i __attribute__((vector_size(16)));   // matches builtin proto

#define HID     768
#define TPB     192            // 6 wave32s, 4 elements per thread
#define NWAVES  (TPB / 32)
#define NRITERS 8

#if defined(__HIP_DEVICE_COMPILE__) && defined(__gfx1250__) && \
    __has_builtin(__builtin_amdgcn_global_load_async_to_lds_b128)
#define USE_ASYNC_LDS 1
typedef v4i __attribute__((address_space(1))) g_v4i;  // global AS int4
typedef v4i __attribute__((address_space(3))) l_v4i;  // LDS AS int4
#else
#define USE_ASYNC_LDS 0
#endif

__global__ __launch_bounds__(TPB) void emb_ln_kernel(
    const int*   __restrict__ input_ids,
    const int*   __restrict__ token_type_ids,
    const int*   __restrict__ word_table,
    const float* __restrict__ word_scale,
    const int*   __restrict__ pos_table,
    const float* __restrict__ pos_scale,
    const int*   __restrict__ type_table,
    const float* __restrict__ type_scale,
    const float* __restrict__ ln_w,
    const float* __restrict__ ln_b,
    float*       __restrict__ out)
{
    __shared__ float s_lnw[HID];
    __shared__ float s_lnb[HID];
    __shared__ float s_red[2 * NWAVES];

    const int s    = blockIdx.x;                    // position id (pos_table row)
    const int row  = blockIdx.y * gridDim.x + s;    // flattened (b, s)
    const int t    = threadIdx.x;
    const int col  = t * 4;
    const int lane = t & 31;                        // wave32
    const int wv   = t >> 5;

    // ---- Stage LayerNorm params into LDS (gfx1250 async DMA path) ----------
#if USE_ASYNC_LDS
    __builtin_amdgcn_global_load_async_to_lds_b128(
        (g_v4i*)(ln_w + col), (l_v4i*)(s_lnw + col), 0, 0);
    __builtin_amdgcn_global_load_async_to_lds_b128(
        (g_v4i*)(ln_b + col), (l_v4i*)(s_lnb + col), 0, 0);
#else
    *(f4*)(s_lnw + col) = *(const f4*)(ln_w + col);
    *(f4*)(s_lnb + col) = *(const f4*)(ln_b + col);
#endif

    // ---- Gather + dequant (all b128, coalesced; row ids are block-uniform) --
    const int   w_id = input_ids[row];
    const int   t_id = token_type_ids[row];
    const float wsc  = word_scale[0];
    const float psc  = pos_scale[0];
    const float tsc  = type_scale[0];

    const i4 wq = *(const i4*)(word_table + (size_t)w_id * HID + col);
    const i4 pq = *(const i4*)(pos_table  + (size_t)s    * HID + col);
    const i4 tq = *(const i4*)(type_table + (size_t)t_id * HID + col);

    float e[4];
#pragma unroll
    for (int i = 0; i < 4; ++i)
        e[i] = (float)wq[i] * wsc + (float)pq[i] * psc + (float)tq[i] * tsc;

    // ---- mean --------------------------------------------------------------
    float sum = (e[0] + e[1]) + (e[2] + e[3]);
#pragma unroll
    for (int m = 16; m >= 1; m >>= 1) sum += __shfl_xor(sum, m, 32);
    if (lane == 0) s_red[wv] = sum;
    __syncthreads();
    float tot = 0.f;
#pragma unroll
    for (int i = 0; i < NWAVES; ++i) tot += s_red[i];
    const float mean = tot * (1.0f / (float)HID);

    // ---- variance (two-pass, faithful to reference) ------------------------
    float ssq = 0.f;
#pragma unroll
    for (int i = 0; i < 4; ++i) { float d = e[i] - mean; ssq += d * d; }
#pragma unroll
    for (int m = 16; m >= 1; m >>= 1) ssq += __shfl_xor(ssq, m, 32);
    if (lane == 0) s_red[NWAVES + wv] = ssq;   // disjoint slots: no extra barrier
    __syncthreads();
    float vtot = 0.f;
#pragma unroll
    for (int i = 0; i < NWAVES; ++i) vtot += s_red[NWAVES + i];
    const float var = vtot * (1.0f / (float)HID);

    // ---- Newton-Raphson sqrt (8 iters), redundant per lane -----------------
    float x = var > 1.0f ? var * 0.5f : 1.0f;
#pragma unroll
    for (int i = 0; i < NRITERS; ++i) x = 0.5f * (x + var / (x + 1e-9f));
    const float inv = 1.0f / (x + 1e-12f);

    // ---- affine + store ----------------------------------------------------
#if USE_ASYNC_LDS
#if __has_builtin(__builtin_amdgcn_s_wait_asynccnt)
    __builtin_amdgcn_s_wait_asynccnt(0);
#else
    asm volatile("s_wait_asynccnt 0" ::: "memory");
#endif
#endif
    // each thread reads exactly the LDS region it loaded itself
    f4 o;
#pragma unroll
    for (int i = 0; i < 4; ++i)
        o[i] = s_lnw[col + i] * ((e[i] - mean) * inv) + s_lnb[col + i];

    // 100 MB write-once stream: NT store keeps word_table resident in L2
    __builtin_nontemporal_store(o, (f4*)(out + (size_t)row * HID + col));
}

extern "C" void kernel_launch(void* const* d_in, const int* in_sizes, int n_in,
                              void* d_out, int out_size, void* d_ws, size_t ws_size,
                              hipStream_t stream) {
    const int*   input_ids  = (const int*)  d_in[0];
    const int*   type_ids   = (const int*)  d_in[1];
    const int*   word_table = (const int*)  d_in[2];
    const float* word_scale = (const float*)d_in[3];
    const int*   pos_table  = (const int*)  d_in[4];
    const float* pos_scale  = (const float*)d_in[5];
    const int*   type_table = (const int*)  d_in[6];
    const float* type_scale = (const float*)d_in[7];
    const float* ln_w       = (const float*)d_in[8];
    const float* ln_b       = (const float*)d_in[9];
    float*       out        = (float*)d_out;

    const int rows = in_sizes[0];          // B*S = 32768
    const int H    = in_sizes[8];          // 768 (kernel hardcodes HID=768)
    const int S    = in_sizes[4] / H;      // 512
    const int B    = rows / S;             // 64
    (void)n_in; (void)out_size; (void)d_ws; (void)ws_size;

    dim3 grid(S, B);
    emb_ln_kernel<<<grid, TPB, 0, stream>>>(
        input_ids, type_ids, word_table, word_scale, pos_table, pos_scale,
        type_table, type_scale, ln_w, ln_b, out);
}